// ODEFuncW_90503550862038
// MI455X (gfx1250) — compile-verified
//
#include <hip/hip_runtime.h>
#include <hip/hip_bf16.h>
#include <math.h>

#define N_NODES 100000
#define N_EDGES 3200000
#define FEAT    128
#define ROWTILES (N_NODES / 16)   // 6250, exact

typedef __attribute__((ext_vector_type(2))) float v2f;
typedef __attribute__((ext_vector_type(8))) float v8f;

// ---------------------------------------------------------------------------
// Kernel 0: zero d_out (it doubles as the ax accumulator for the SpMM)
// ---------------------------------------------------------------------------
__global__ void odefw_zero_kernel(float4* __restrict__ p, int n4) {
    int i = blockIdx.x * blockDim.x + threadIdx.x;
    if (i < n4) p[i] = make_float4(0.f, 0.f, 0.f, 0.f);
}

// ---------------------------------------------------------------------------
// Kernel 1: w_eff[i][j] = sum_k w[i][k] * clamp(d[k],0,1) * w[j][k]
// 128x128 output, trivial FLOP count -> simple per-element kernel.
// ---------------------------------------------------------------------------
__global__ void odefw_weff_kernel(const float* __restrict__ w,
                                  const float* __restrict__ d,
                                  float* __restrict__ weff) {
    int idx = blockIdx.x * blockDim.x + threadIdx.x;   // 0 .. 16383
    int i = idx >> 7;
    int j = idx & 127;
    float acc = 0.f;
#pragma unroll 4
    for (int k = 0; k < FEAT; ++k) {
        float dc = d[k];
        dc = dc < 0.f ? 0.f : (dc > 1.f ? 1.f : dc);
        acc = fmaf(w[i * FEAT + k] * dc, w[j * FEAT + k], acc);
    }
    weff[idx] = acc;
}

// ---------------------------------------------------------------------------
// Kernel 2: segmented SpMM  ax[r,:] += val_e * x[c_e,:]  (rows sorted)
// One wave per 256-edge strip. Lane l owns columns [4l, 4l+4).
// Register accumulation while the row is unchanged; atomic flush at
// row boundaries (and strip ends) only.
// ---------------------------------------------------------------------------
#define EDGES_PER_WAVE 256

__device__ __forceinline__ void odefw_flush(float* __restrict__ ax, int row,
                                            int lane, const float4& acc) {
    float* dst = ax + row * FEAT + lane * 4;
    unsafeAtomicAdd(dst + 0, acc.x);
    unsafeAtomicAdd(dst + 1, acc.y);
    unsafeAtomicAdd(dst + 2, acc.z);
    unsafeAtomicAdd(dst + 3, acc.w);
}

__global__ __launch_bounds__(256) void odefw_spmm_kernel(
        const float* __restrict__ x,
        const float* __restrict__ vals,
        const int*   __restrict__ rows,
        const int*   __restrict__ cols,
        float*       __restrict__ ax) {
    const int lane = threadIdx.x & 31;
    const int wave = (blockIdx.x * blockDim.x + threadIdx.x) >> 5;
    int e0 = wave * EDGES_PER_WAVE;
    if (e0 >= N_EDGES) return;
    int e1 = e0 + EDGES_PER_WAVE;
    if (e1 > N_EDGES) e1 = N_EDGES;

    int cur_row = rows[e0];
    float4 acc = make_float4(0.f, 0.f, 0.f, 0.f);

    for (int eb = e0; eb < e1; eb += 32) {
        int n = e1 - eb;
        if (n > 32) n = 32;
        int   eg = eb + lane;
        bool  ok = eg < N_EDGES;
        int   rl = ok ? rows[eg] : 0;
        int   cl = ok ? cols[eg] : 0;
        float vl = ok ? vals[eg] : 0.f;
        for (int j = 0; j < n; ++j) {
            int r = __shfl(rl, j, 32);           // uniform per wave
            if (r != cur_row) {
                odefw_flush(ax, cur_row, lane, acc);
                acc = make_float4(0.f, 0.f, 0.f, 0.f);
                cur_row = r;
            }
            int   c = __shfl(cl, j, 32);
            float v = __shfl(vl, j, 32);
            const float4 xv =
                *(const float4*)(x + c * FEAT + lane * 4);  // coalesced 512B
            acc.x = fmaf(v, xv.x, acc.x);
            acc.y = fmaf(v, xv.y, acc.y);
            acc.z = fmaf(v, xv.z, acc.z);
            acc.w = fmaf(v, xv.w, acc.w);
        }
    }
    odefw_flush(ax, cur_row, lane, acc);
}

// ---------------------------------------------------------------------------
// Kernel 3: fused  xw = x @ w_eff  (V_WMMA_F32_16X16X4_F32)  +  epilogue
//   f = 0.5*sigmoid(alpha)*(ax - x) + xw - x + x0
// out[] contains ax on entry, overwritten with f (each element owned by
// exactly one thread: read-before-write, no cross-thread aliasing).
//
// Block = 256 threads = 8 waves; wave wv computes row-tile blockIdx*8+wv
// (16 rows x all 128 cols = 8 WMMA accumulators). K loop: 32 steps of K=4.
// w_eff is staged in LDS pre-swizzled into exact B-fragment layout:
//   lds[((ks*8 + t)*32 + lane)*2 + c] = weff[(4*ks + 2*(lane>>4) + c)][16*t + (lane&15)]
// so each B fragment is one contiguous conflict-free 8B ds load per lane.
//
// f32 WMMA fragment layout (per CDNA5 ISA):
//   A 16x4 : lanes 0-15 -> M=0..15 ; VGPR0 = K=0 (lanes<16) / K=2 (lanes>=16)
//   C 16x16: VGPR v, lane l -> row v + 8*(l>>4), col (l&15)
// ---------------------------------------------------------------------------
__global__ __launch_bounds__(256) void odefw_fused_kernel(
        const float* __restrict__ x,
        const float* __restrict__ x0,
        const float* __restrict__ alpha,
        const float* __restrict__ weff,
        float*       __restrict__ out /* ax in, f out */) {
    __shared__ float ldsb[32 * 8 * 32 * 2];   // 64 KB, B fragments
    const int tid  = threadIdx.x;
    const int lane = tid & 31;
    const int wv   = tid >> 5;

    // cooperative swizzled preload of w_eff into B-fragment order
    for (int idx = tid; idx < 32 * 8 * 32 * 2; idx += 256) {
        int c  = idx & 1;
        int l  = (idx >> 1) & 31;
        int t  = (idx >> 6) & 7;
        int ks = idx >> 9;
        int rw = 4 * ks + 2 * (l >> 4) + c;   // K index
        int cw = 16 * t + (l & 15);           // N index
        ldsb[idx] = weff[rw * FEAT + cw];
    }
    __syncthreads();

    const int rowTile = blockIdx.x * 8 + wv;
    if (rowTile >= ROWTILES) return;
    const int r0 = rowTile * 16;

    v8f acc[8] = {};

    const int abase = (r0 + (lane & 15)) * FEAT + 2 * (lane >> 4);
    for (int ks = 0; ks < 32; ++ks) {
        v2f a = *(const v2f*)(x + abase + 4 * ks);       // A fragment
        const float* bp = &ldsb[ks * 512 + lane * 2];
#pragma unroll
        for (int t = 0; t < 8; ++t) {
            v2f b = *(const v2f*)(bp + t * 64);          // B fragment
            acc[t] = __builtin_amdgcn_wmma_f32_16x16x4_f32(
                false, a, false, b, (short)0, acc[t], false, false);
        }
    }

    // fused epilogue
    const int rOff = 8 * (lane >> 4);
    const int col  = lane & 15;
#pragma unroll
    for (int v = 0; v < 8; ++v) {
        int row = r0 + v + rOff;
        float s = 1.0f / (1.0f + __expf(-alpha[row]));   // sigmoid
        int base = row * FEAT + col;
#pragma unroll
        for (int t = 0; t < 8; ++t) {
            int idx = base + 16 * t;
            float xv  = x[idx];
            float axv = out[idx];                        // ax (own element)
            float x0v = x0[idx];
            out[idx] = fmaf(0.5f * s, axv - xv, acc[t][v] - xv + x0v);
        }
    }
}

// ---------------------------------------------------------------------------
extern "C" void kernel_launch(void* const* d_in, const int* in_sizes, int n_in,
                              void* d_out, int out_size, void* d_ws, size_t ws_size,
                              hipStream_t stream) {
    const float* x           = (const float*)d_in[0];
    const float* x0          = (const float*)d_in[1];
    const float* adj_vals    = (const float*)d_in[2];
    const float* alpha_train = (const float*)d_in[3];
    const float* w           = (const float*)d_in[4];
    const float* d           = (const float*)d_in[5];
    const int*   adj_rows    = (const int*)d_in[6];
    const int*   adj_cols    = (const int*)d_in[7];
    float* out  = (float*)d_out;
    float* weff = (float*)d_ws;   // 128*128*4 = 64 KB of scratch

    // 1) zero ax accumulator (== d_out)
    int n4 = N_NODES * FEAT / 4;                       // 3.2M float4
    odefw_zero_kernel<<<(n4 + 255) / 256, 256, 0, stream>>>((float4*)out, n4);

    // 2) w_eff
    odefw_weff_kernel<<<(FEAT * FEAT) / 256, 256, 0, stream>>>(w, d, weff);

    // 3) segmented SpMM -> ax (in d_out)
    int nWaves  = (N_EDGES + EDGES_PER_WAVE - 1) / EDGES_PER_WAVE;  // 12500
    int nThr    = nWaves * 32;
    odefw_spmm_kernel<<<(nThr + 255) / 256, 256, 0, stream>>>(
        x, adj_vals, adj_rows, adj_cols, out);

    // 4) WMMA GEMM + fused epilogue (overwrites ax with f)
    int nBlocks = (ROWTILES + 7) / 8;                  // 782
    odefw_fused_kernel<<<nBlocks, 256, 0, stream>>>(x, x0, alpha_train, weff, out);
}